// HetroGAT_1803886264482
// MI455X (gfx1250) — compile-verified
//
#include <hip/hip_runtime.h>
#include <hip/hip_bf16.h>

// ---------------------------------------------------------------------------
// Hetero-GAT for MI455X (gfx1250, wave32).
//  - All N x K @ K x 128 GEMMs via V_WMMA_F32_16X16X4_F32 (full fp32 path).
//  - Edge segment-softmax: 2 edge passes (atomicMax of order-encoded floats,
//    then fused exp + denom atomicAdd + unnormalized message scatter) and one
//    node pass for normalization + bias + relation accumulation.
//  - All hot per-node arrays (<=25.6MB each) live in workspace and fit in the
//    192MB L2, so random edge gathers/atomics stay on-chip.
// ---------------------------------------------------------------------------

#define N_NODES 50000
#define E_EDGES 800000
#define R_REL   3
#define IN_DIM  32
#define HID     128
#define HEADS   16
#define DOUT    8
#define LAYERS  4

typedef __attribute__((ext_vector_type(2))) float v2f;
typedef __attribute__((ext_vector_type(8))) float v8f;

// ---------------- order-preserving float <-> uint encoding -----------------
__device__ __forceinline__ unsigned fenc(float f) {
    unsigned u = __float_as_uint(f);
    return (u & 0x80000000u) ? ~u : (u | 0x80000000u);
}
__device__ __forceinline__ float fdec(unsigned u) {
    unsigned v = (u & 0x80000000u) ? (u & 0x7FFFFFFFu) : ~u;
    return __uint_as_float(v);
}
__device__ __forceinline__ float lrelu(float x, float ns) {
    return x > 0.0f ? x : ns * x;
}

// ---------------------------------------------------------------------------
// C[M x 128] = act(A[M x K] @ W[K x 128] + bias)
// One wave = one 16x16 tile. blockDim = 256 (8 waves -> 8 column tiles),
// gridDim.x = M/16. K must be a multiple of 4 (32 or 128 here).
//
// Fragment layout per ISA (32-bit A 16x4): lanes 0-15 carry M=lid at K={kk,kk+1},
// lanes 16-31 carry M=lid at K={kk+2,kk+3}. B mirrors with N=lid. D: VGPR i of
// lanes 0-15 is row m0+i, lanes 16-31 row m0+8+i.
// ---------------------------------------------------------------------------
__global__ void gemm128_wmma(const float* __restrict__ A, int K,
                             const float* __restrict__ W,
                             const float* __restrict__ bias,
                             float* __restrict__ C, int relu)
{
    const int lane = threadIdx.x & 31;
    const int wave = threadIdx.x >> 5;   // column tile 0..7
    const int half = lane >> 4;          // 0: K pair {0,1}, 1: K pair {2,3}
    const int lid  = lane & 15;
    const int m0   = blockIdx.x * 16;
    const int col  = wave * 16 + lid;

    const float* __restrict__ arow = A + (size_t)(m0 + lid) * K;

    v8f acc = {};
    for (int kk = 0; kk < K; kk += 4) {
        const int ka = kk + half * 2;
        v2f a = *(const v2f*)(arow + ka);          // A[m][ka], A[m][ka+1]
        v2f b;
        b.x = W[(size_t)(ka + 0) * HID + col];     // B[ka  ][n]
        b.y = W[(size_t)(ka + 1) * HID + col];     // B[ka+1][n]
        acc = __builtin_amdgcn_wmma_f32_16x16x4_f32(
                  false, a, false, b, (short)0, acc, false, false);
    }

    const float bv = bias ? bias[col] : 0.0f;
#pragma unroll
    for (int i = 0; i < 8; ++i) {
        float v = acc[i] + bv;
        if (relu) v = fmaxf(v, 0.0f);
        C[(size_t)(m0 + half * 8 + i) * HID + col] = v;
    }
}

// el/er: per (node, head) attention logit halves. t = n*HEADS + h.
__global__ void attn_el_er(const float* __restrict__ hp,
                           const float* __restrict__ al,
                           const float* __restrict__ ar,
                           float* __restrict__ el, float* __restrict__ er)
{
    const int t = blockIdx.x * blockDim.x + threadIdx.x;
    if (t >= N_NODES * HEADS) return;
    const int h = t & (HEADS - 1);
    const float* __restrict__ v  = hp + (size_t)(t >> 4) * HID + h * DOUT;
    const float* __restrict__ pa = al + h * DOUT;
    const float* __restrict__ pb = ar + h * DOUT;
    float sl = 0.0f, sr = 0.0f;
#pragma unroll
    for (int d = 0; d < DOUT; ++d) { const float x = v[d]; sl += x * pa[d]; sr += x * pb[d]; }
    el[t] = sl;
    er[t] = sr;
}

__global__ void fill_zero_u32(unsigned* __restrict__ p, size_t n)
{
    const size_t i = (size_t)blockIdx.x * blockDim.x + threadIdx.x;
    if (i < n) p[i] = 0u;
}

// Pass 1: per (edge, head) running max over incoming edges of each dst.
__global__ void edge_max(const int* __restrict__ src, const int* __restrict__ dst,
                         const float* __restrict__ el, const float* __restrict__ er,
                         unsigned* __restrict__ menc)
{
    const int t = blockIdx.x * blockDim.x + threadIdx.x;
    if (t >= E_EDGES * HEADS) return;
    const int e = t >> 4, h = t & (HEADS - 1);
    const int s = src[e], d = dst[e];
    const float ev = lrelu(el[s * HEADS + h] + er[d * HEADS + h], 0.2f);
    atomicMax(menc + d * HEADS + h, fenc(ev));
}

// Pass 2: ex = exp(e - m); denom += ex; tmp[dst] += hp[src] * ex (unnormalized).
__global__ void edge_accum(const int* __restrict__ src, const int* __restrict__ dst,
                           const float* __restrict__ el, const float* __restrict__ er,
                           const unsigned* __restrict__ menc,
                           const float* __restrict__ hp,
                           float* __restrict__ ssum, float* __restrict__ tmp)
{
    const int t = blockIdx.x * blockDim.x + threadIdx.x;
    if (t >= E_EDGES * HEADS) return;
    const int e = t >> 4, h = t & (HEADS - 1);
    const int s = src[e], d = dst[e];
    const float ev = lrelu(el[s * HEADS + h] + er[d * HEADS + h], 0.2f);
    const float m  = fdec(menc[d * HEADS + h]);
    const float ex = __expf(ev - m);
    atomicAdd(ssum + d * HEADS + h, ex);
    const float* __restrict__ hv = hp  + (size_t)s * HID + h * DOUT;
    float* __restrict__       ov = tmp + (size_t)d * HID + h * DOUT;
#pragma unroll
    for (int dd = 0; dd < DOUT; ++dd) atomicAdd(ov + dd, hv[dd] * ex);
}

// Node pass: acc += tmp/denom + bias (per relation). t = n*HEADS + h.
__global__ void node_norm_acc(const float* __restrict__ ssum,
                              const float* __restrict__ tmp,
                              const float* __restrict__ b,
                              float* __restrict__ acc)
{
    const int t = blockIdx.x * blockDim.x + threadIdx.x;
    if (t >= N_NODES * HEADS) return;
    const int h = t & (HEADS - 1);
    const float inv = 1.0f / fmaxf(ssum[t], 1e-38f);
    const float* __restrict__ tv = tmp + (size_t)(t >> 4) * HID + h * DOUT;
    float* __restrict__       av = acc + (size_t)(t >> 4) * HID + h * DOUT;
    const float* __restrict__ bv = b + h * DOUT;
#pragma unroll
    for (int d = 0; d < DOUT; ++d) av[d] += tv[d] * inv + bv[d];
}

// h = leaky_relu(acc, 0.01) + h  (residual, in place)
__global__ void layer_update(const float* __restrict__ acc, float* __restrict__ h)
{
    const size_t i = (size_t)blockIdx.x * blockDim.x + threadIdx.x;
    if (i >= (size_t)N_NODES * HID) return;
    h[i] += lrelu(acc[i], 0.01f);
}

// out[n] = dot(t[n,:], W2[:,0]) + b2
__global__ void dec_out(const float* __restrict__ t, const float* __restrict__ w2,
                        const float* __restrict__ b2, float* __restrict__ out)
{
    const int n = blockIdx.x * blockDim.x + threadIdx.x;
    if (n >= N_NODES) return;
    const float* __restrict__ v = t + (size_t)n * HID;
    float s = 0.0f;
#pragma unroll 4
    for (int k = 0; k < HID; ++k) s += v[k] * w2[k];
    out[n] = s + b2[0];
}

extern "C" void kernel_launch(void* const* d_in, const int* in_sizes, int n_in,
                              void* d_out, int out_size, void* d_ws, size_t ws_size,
                              hipStream_t stream)
{
    (void)in_sizes; (void)n_in; (void)out_size; (void)ws_size;

    // -------- inputs (setup_inputs dict order, params in insertion order) ---
    const float* x       = (const float*)d_in[0];
    const int*   src_all = (const int*)d_in[1];   // [R, E]
    const int*   dst_all = (const int*)d_in[2];   // [R, E]
    int p = 3;
    const float* eW1 = (const float*)d_in[p++];
    const float* eb1 = (const float*)d_in[p++];
    const float* eW2 = (const float*)d_in[p++];
    const float* eb2 = (const float*)d_in[p++];
    const float *gW[LAYERS][R_REL], *gal[LAYERS][R_REL], *gar[LAYERS][R_REL], *gb[LAYERS][R_REL];
    for (int l = 0; l < LAYERS; ++l)
        for (int r = 0; r < R_REL; ++r) {
            gW [l][r] = (const float*)d_in[p++];
            gal[l][r] = (const float*)d_in[p++];
            gar[l][r] = (const float*)d_in[p++];
            gb [l][r] = (const float*)d_in[p++];
        }
    const float* dW1 = (const float*)d_in[p++];
    const float* db1 = (const float*)d_in[p++];
    const float* dW2 = (const float*)d_in[p++];
    const float* db2 = (const float*)d_in[p++];

    // -------- workspace layout (all fp32) ----------------------------------
    // h, hp, acc: N*128 each; then contiguous [menc | ssum | tmp] so one
    // zero-fill resets per-relation state; then el, er.
    char* ws = (char*)d_ws;
    const size_t NH  = (size_t)N_NODES * HID;      // 6.4M elems
    const size_t NHe = (size_t)N_NODES * HEADS;    // 0.8M elems
    float*    h    = (float*)ws;                 ws += NH  * 4;
    float*    hp   = (float*)ws;                 ws += NH  * 4;
    float*    acc  = (float*)ws;                 ws += NH  * 4;
    unsigned* menc = (unsigned*)ws;              ws += NHe * 4;
    float*    ssum = (float*)ws;                 ws += NHe * 4;
    float*    tmp  = (float*)ws;                 ws += NH  * 4;
    float*    el   = (float*)ws;                 ws += NHe * 4;
    float*    er   = (float*)ws;                 ws += NHe * 4;
    float*    out  = (float*)d_out;

    const int TB = 256;
    const int gGemm = N_NODES / 16;                       // 3125 (exact)
    const int gNH   = (int)((NH  + TB - 1) / TB);         // 25000
    const int gNHe  = (int)((NHe + TB - 1) / TB);         // 3125
    const int gEH   = (E_EDGES * HEADS + TB - 1) / TB;    // 50000
    const size_t relZero = NHe + NHe + NH;                // menc+ssum+tmp dwords
    const int gRZ   = (int)((relZero + TB - 1) / TB);
    const int gN    = (N_NODES + TB - 1) / TB;

    // -------- embed MLP: h = relu(x@W1+b1)@W2 + b2 --------------------------
    gemm128_wmma<<<gGemm, TB, 0, stream>>>(x,  IN_DIM, eW1, eb1, hp, 1);
    gemm128_wmma<<<gGemm, TB, 0, stream>>>(hp, HID,    eW2, eb2, h,  0);

    // -------- GAT layers ----------------------------------------------------
    for (int l = 0; l < LAYERS; ++l) {
        fill_zero_u32<<<gNH, TB, 0, stream>>>((unsigned*)acc, NH);
        for (int r = 0; r < R_REL; ++r) {
            const int* src = src_all + (size_t)r * E_EDGES;
            const int* dst = dst_all + (size_t)r * E_EDGES;
            // hp = h @ W  (WMMA), then per-head logit halves
            gemm128_wmma<<<gGemm, TB, 0, stream>>>(h, HID, gW[l][r], nullptr, hp, 0);
            attn_el_er<<<gNHe, TB, 0, stream>>>(hp, gal[l][r], gar[l][r], el, er);
            // reset m (enc 0 == below any finite), denom, message accumulator
            fill_zero_u32<<<gRZ, TB, 0, stream>>>(menc, relZero);
            edge_max  <<<gEH, TB, 0, stream>>>(src, dst, el, er, menc);
            edge_accum<<<gEH, TB, 0, stream>>>(src, dst, el, er, menc, hp, ssum, tmp);
            node_norm_acc<<<gNHe, TB, 0, stream>>>(ssum, tmp, gb[l][r], acc);
        }
        layer_update<<<gNH, TB, 0, stream>>>(acc, h);
    }

    // -------- decoder -------------------------------------------------------
    gemm128_wmma<<<gGemm, TB, 0, stream>>>(h, HID, dW1, db1, hp, 1);
    dec_out<<<gN, TB, 0, stream>>>(hp, dW2, db2, out);
}